// GaborConv2d_47467978555766
// MI455X (gfx1250) — compile-verified
//
#include <hip/hip_runtime.h>
#include <hip/hip_bf16.h>

typedef __attribute__((ext_vector_type(2))) float v2f;
typedef __attribute__((ext_vector_type(8))) float v8f;

#define OUT_HW   510
#define OUT_PIX  (OUT_HW * OUT_HW)   // 260100
#define IN_HW    512
#define IN_CHPIX (IN_HW * IN_HW)     // 262144
#define K_RAW    27                  // 3 channels * 3x3 taps
#define K_PAD    28                  // pad to multiple of 4 for 16x16x4 WMMA
#define N_BATCH  16
#define N_OUT_CH 32

// ---------------------------------------------------------------------------
// Kernel 1: build the Gabor weight matrix, stored K-padded as Wt[o][k],
// o in [0,32), k = ci*9 + ky*3 + kx in [0,27), Wt[o][27] = 0 padding.
// Mirrors the JAX reference exactly (PI = 3.14, linspace(-1,2,3) grid).
// ---------------------------------------------------------------------------
__global__ void gabor_weights_kernel(const float* __restrict__ freq,
                                     const float* __restrict__ theta,
                                     const float* __restrict__ psi,
                                     const float* __restrict__ sigma,
                                     float* __restrict__ Wt) {
    int idx = blockIdx.x * blockDim.x + threadIdx.x;
    if (idx < N_OUT_CH) Wt[idx * K_PAD + K_RAW] = 0.0f;   // zero the K pad
    if (idx >= N_OUT_CH * K_RAW) return;

    int o  = idx / K_RAW;
    int k  = idx - o * K_RAW;
    int ci = k / 9;
    int r  = k - ci * 9;
    int ky = r / 3;
    int kx = r - ky * 3;

    // linspace(-x0+1, x0, 3) with x0 = ceil(3/2) = 2  ->  {-1, 0.5, 2}
    float xg = (kx == 0) ? -1.0f : ((kx == 1) ? 0.5f : 2.0f);
    float yg = (ky == 0) ? -1.0f : ((ky == 1) ? 0.5f : 2.0f);

    int   oi = o * 3 + ci;
    float f  = freq[oi];
    float th = theta[oi];
    float p  = psi[oi];
    float s  = sigma[oi];

    float st, ct;
    sincosf(th, &st, &ct);
    float rotx =  xg * ct + yg * st;
    float roty = -xg * st + yg * ct;
    float sp   = s + 0.001f;
    float g    = expf(-0.5f * (rotx * rotx + roty * roty) / (sp * sp));
    g *= cosf(f * rotx + p);
    g /= (2.0f * 3.14f * s * s);

    Wt[o * K_PAD + k] = g;
}

// ---------------------------------------------------------------------------
// Kernel 2: implicit-GEMM conv via V_WMMA_F32_16X16X4_F32.
// One block (8 wave32s) per output row; each wave handles 4 tiles of
// M=32 out-channels x N=16 pixels. Weights + im2col tap offsets are hoisted
// out of the tile loop; per tile only cheap u32 offset adds remain.
// ---------------------------------------------------------------------------
__global__ void __launch_bounds__(256)
gabor_conv_wmma_kernel(const float* __restrict__ in,
                       const float* __restrict__ Wt,
                       float* __restrict__ out) {
    const int lane = threadIdx.x & 31;
    const int wave = threadIdx.x >> 5;
    const int l16  = lane & 15;
    const int hi   = lane >> 4;          // selects K-pair within a 4-wide step

    const int row = blockIdx.x;          // 0 .. 16*510-1
    const int y   = row % OUT_HW;
    const int b   = row / OUT_HW;
    const int xw  = wave * 64;           // this wave's 4-tile base column

    const float* __restrict__ inb  = in  + (size_t)b * 3 * IN_CHPIX;
    float* __restrict__       outb = out + (size_t)b * N_OUT_CH * OUT_PIX;

    // ---- A fragments: weights for o = l16 (c0) and o = 16+l16 (c1) --------
    // 16x4 f32 A layout: VGPR v holds K = 4s + 2*hi + v for lane row M.
    const float* W0 = Wt + l16 * K_PAD + 2 * hi;
    const float* W1 = Wt + (l16 + 16) * K_PAD + 2 * hi;
    v2f a0[7], a1[7];
#pragma unroll
    for (int s = 0; s < 7; ++s) {
        a0[s][0] = W0[4 * s];
        a0[s][1] = W0[4 * s + 1];
        a1[s][0] = W1[4 * s];
        a1[s][1] = W1[4 * s + 1];
    }

    // ---- hoisted im2col tap offsets (u32 floats into inb, span < 3MB) -----
    // tap k = 4s + 2*hi + v -> (ci,ky,kx); k==27 is the K pad (alias to k=0,
    // value zeroed per tile). Offsets depend only on wave/lane, not tile.
    unsigned tb[7][2];
#pragma unroll
    for (int s = 0; s < 7; ++s) {
#pragma unroll
        for (int v = 0; v < 2; ++v) {
            int k  = 4 * s + 2 * hi + v;
            int kc = (k < K_RAW) ? k : 0;
            int ci = kc / 9;
            int r  = kc - ci * 9;
            int ky = r / 3;
            int kx = r - ky * 3;
            tb[s][v] = (unsigned)((ci * IN_HW + (y + ky)) * IN_HW + xw + l16 + kx);
        }
    }

    // ---- 4 tiles per wave; last tile of the row starts at 494 (overlap) ---
#pragma unroll
    for (int t = 0; t < 4; ++t) {
        int xs = xw + 16 * t;
        if (xs > OUT_HW - 16) xs = OUT_HW - 16;      // 494: overlap, no masking
        const unsigned d = (unsigned)(xs - xw);      // uniform tile delta (floats)

        // B fragments: 14 gathers at hoisted offsets + tile delta
        v2f bf[7];
#pragma unroll
        for (int s = 0; s < 7; ++s) {
#pragma unroll
            for (int v = 0; v < 2; ++v) {
                float val = inb[tb[s][v] + d];
                if (4 * s + 2 * hi + v >= K_RAW) val = 0.0f;  // K pad lane
                bf[s][v] = val;
            }
        }

        // 7 chained f32 WMMAs per accumulator (K = 0..27)
        v8f c0 = {};
        v8f c1 = {};
#pragma unroll
        for (int s = 0; s < 7; ++s) {
            c0 = __builtin_amdgcn_wmma_f32_16x16x4_f32(
                     false, a0[s], false, bf[s], (short)0, c0, false, false);
            c1 = __builtin_amdgcn_wmma_f32_16x16x4_f32(
                     false, a1[s], false, bf[s], (short)0, c1, false, false);
        }

        // stores: C/D layout row M = r + 8*hi, col N = l16.
        // r*OUT_PIX folds into the 24-bit store immediate (max 7.3MB, fits).
        unsigned p0 = (unsigned)(hi * 8 * OUT_PIX + y * OUT_HW + xs + l16);
        unsigned p1 = p0 + 16u * OUT_PIX;
#pragma unroll
        for (int r = 0; r < 8; ++r) {
            outb[p0 + r * OUT_PIX] = c0[r];
            outb[p1 + r * OUT_PIX] = c1[r];
        }
    }
}

// ---------------------------------------------------------------------------
extern "C" void kernel_launch(void* const* d_in, const int* in_sizes, int n_in,
                              void* d_out, int out_size, void* d_ws, size_t ws_size,
                              hipStream_t stream) {
    (void)in_sizes; (void)n_in; (void)out_size; (void)ws_size;

    const float* in    = (const float*)d_in[0];  // (16,3,512,512)
    const float* freq  = (const float*)d_in[1];  // (32,3)
    const float* theta = (const float*)d_in[2];
    const float* psi   = (const float*)d_in[3];
    const float* sigma = (const float*)d_in[4];
    float*       out   = (float*)d_out;          // (16,32,510,510)
    float*       Wt    = (float*)d_ws;           // 32*28 floats scratch

    gabor_weights_kernel<<<4, 256, 0, stream>>>(freq, theta, psi, sigma, Wt);

    const int blocks = N_BATCH * OUT_HW;         // one block per output row
    gabor_conv_wmma_kernel<<<blocks, 256, 0, stream>>>(in, Wt, out);
}